// Router_4896262717685
// MI455X (gfx1250) — compile-verified
//
#include <hip/hip_runtime.h>
#include <hip/hip_bf16.h>

// Problem constants (match reference)
#define NTOK   4096      // B*T
#define DEMB   1024
#define NEXP   8
#define CAP    1280
#define EC     (NEXP * CAP)           // 10240
#define CB_OFF 8                      // used_capacity occupies out[0..7]
#define MASK_OFF (8 + (long)NTOK * EC)  // 41943048

typedef __attribute__((ext_vector_type(2))) float v2f;
typedef __attribute__((ext_vector_type(8))) float v8f;

static __device__ __forceinline__ unsigned ballot32(bool p) {
    return __builtin_amdgcn_ballot_w32(p);
}

// ---------------------------------------------------------------------------
// Kernel 1: zero the entire output (335 MB) with b128 stores. This is the
// bandwidth-dominant part of the whole problem (~14.4 us at 23.3 TB/s).
// ---------------------------------------------------------------------------
__global__ void zero_out_kernel(float* __restrict__ out, long n_elems) {
    long tid    = (long)blockIdx.x * blockDim.x + threadIdx.x;
    long stride = (long)gridDim.x * blockDim.x;
    long n4 = n_elems >> 2;
    float4 z = make_float4(0.f, 0.f, 0.f, 0.f);
    float4* out4 = reinterpret_cast<float4*>(out);
    for (long i = tid; i < n4; i += stride) out4[i] = z;
    // tail (n_elems % 4) — out_size is divisible by 4 here, but be safe
    for (long j = (n4 << 2) + tid; j < n_elems; j += stride) out[j] = 0.f;
}

// ---------------------------------------------------------------------------
// Kernel 2: logits = x @ w_g^T via V_WMMA_F32_16X16X4_F32, then per-token
// top-2 + 2-way softmax. One wave (32 lanes) per 16-token tile.
//
// A (16x4, f32): lane<16 holds M=lane,K={k,k+1}; lane>=16 holds M=lane-16,
//                K={k+2,k+3}  (ISA 7.12.2 32-bit A layout)
// B (4x16, f32): mirrored — lane<16 holds N=lane rows K={k,k+1}; lane>=16
//                holds N=lane-16 rows K={k+2,k+3}. Experts 8..15 padded to 0.
// C/D (16x16):   vgpr i: lanes 0-15 -> M=i, lanes 16-31 -> M=i+8, N=lane%16.
// ---------------------------------------------------------------------------
__global__ __launch_bounds__(32)
void logits_top2_kernel(const float* __restrict__ x,
                        const float* __restrict__ wg,
                        int* __restrict__ e0, int* __restrict__ e1,
                        float* __restrict__ p0, float* __restrict__ p1) {
    const int lane = threadIdx.x;           // 0..31
    const int tile = blockIdx.x;            // 0..255 (4096/16 tiles)
    const int m    = lane & 15;             // token-in-tile / expert column
    const int koff = (lane >> 4) << 1;      // 0 or 2

    const float* xrow = x  + (long)(tile * 16 + m) * DEMB;
    const int    esafe = (m < NEXP) ? m : (NEXP - 1);
    const float* wrow  = wg + (long)esafe * DEMB;
    const float  wscale = (m < NEXP) ? 1.0f : 0.0f;   // zero-pad experts 8..15

    v8f acc = {};
    for (int k = 0; k < DEMB; k += 4) {
        v2f a = *reinterpret_cast<const v2f*>(xrow + k + koff);
        v2f b = *reinterpret_cast<const v2f*>(wrow + k + koff);
        b = b * wscale;
        acc = __builtin_amdgcn_wmma_f32_16x16x4_f32(
            /*neg_a=*/false, a, /*neg_b=*/false, b,
            /*c_mod=*/(short)0, acc, /*reuse_a=*/false, /*reuse_b=*/false);
    }

    // Transpose the 16x16 tile through LDS: lds[token][expert]
    __shared__ float lds[16][16];
    const int rowbase = (lane >> 4) * 8;
    #pragma unroll
    for (int i = 0; i < 8; ++i) lds[rowbase + i][m] = acc[i];
    __syncthreads();

    if (lane < 16) {
        // top-2 over 8 logits; strict '>' keeps the lower index on ties,
        // matching jax.lax.top_k tie-breaking.
        float v0 = -3.402823466e38f, v1 = -3.402823466e38f;
        int   i0 = 0, i1 = 0;
        #pragma unroll
        for (int e = 0; e < NEXP; ++e) {
            float v = lds[lane][e];
            if (v > v0)      { v1 = v0; i1 = i0; v0 = v; i0 = e; }
            else if (v > v1) { v1 = v;  i1 = e; }
        }
        // softmax over the two selected logits (max-subtracted, like jax)
        float t  = expf(v1 - v0);
        float s  = 1.0f + t;
        int n = tile * 16 + lane;
        e0[n] = i0;  e1[n] = i1;
        p0[n] = 1.0f / s;
        p1[n] = t / s;
    }
}

// ---------------------------------------------------------------------------
// Kernel 3: per-row expert histograms. Row r of 256 covers 32 entries of the
// k-major sequence (r<128 -> first choices, r>=128 -> second choices).
// ---------------------------------------------------------------------------
__global__ __launch_bounds__(32)
void row_hist_kernel(const int* __restrict__ e0, const int* __restrict__ e1,
                     int* __restrict__ cnt) {
    const int lane = threadIdx.x;
    const int r    = blockIdx.x;            // 0..255
    const int k    = r >> 7;
    const int n    = (r & 127) * 32 + lane;
    const int e    = k ? e1[n] : e0[n];

    int myCnt = 0;
    #pragma unroll
    for (int q = 0; q < NEXP; ++q) {
        unsigned bal = ballot32(e == q);
        if (lane == q) myCnt = __popc(bal);
    }
    if (lane < NEXP) cnt[r * NEXP + lane] = myCnt;
}

// ---------------------------------------------------------------------------
// Kernel 4: single-block exclusive scan of the 256x8 histogram (Hillis-
// Steele in LDS) -> per-row base offsets; also emits used_capacity.
// ---------------------------------------------------------------------------
__global__ __launch_bounds__(256)
void scan_kernel(const int* __restrict__ cnt, int* __restrict__ base,
                 float* __restrict__ out) {
    __shared__ int s[256][NEXP];
    const int r = threadIdx.x;              // 0..255
    int v[NEXP];
    #pragma unroll
    for (int e = 0; e < NEXP; ++e) { v[e] = cnt[r * NEXP + e]; s[r][e] = v[e]; }
    __syncthreads();

    for (int off = 1; off < 256; off <<= 1) {
        int t[NEXP];
        if (r >= off) {
            #pragma unroll
            for (int e = 0; e < NEXP; ++e) t[e] = s[r - off][e];
        }
        __syncthreads();
        if (r >= off) {
            #pragma unroll
            for (int e = 0; e < NEXP; ++e) s[r][e] += t[e];
        }
        __syncthreads();
    }

    #pragma unroll
    for (int e = 0; e < NEXP; ++e) base[r * NEXP + e] = s[r][e] - v[e];
    if (r < NEXP) {
        int total = s[255][r];
        out[r] = (float)(total < CAP ? total : CAP);   // used_capacity
    }
}

// ---------------------------------------------------------------------------
// Kernel 5: scatter. Each wave handles one 32-entry row: intra-row exclusive
// prefix via ballot + scanned base gives the global k-major rank; entries
// with rank < CAP write cb_weight (prob) and sec_mask (1.0).
// ---------------------------------------------------------------------------
__global__ __launch_bounds__(32)
void scatter_kernel(const int* __restrict__ e0, const int* __restrict__ e1,
                    const float* __restrict__ p0, const float* __restrict__ p1,
                    const int* __restrict__ base, float* __restrict__ out) {
    const int lane = threadIdx.x;
    const int r    = blockIdx.x;            // 0..255
    const int k    = r >> 7;
    const int n    = (r & 127) * 32 + lane;
    const int e    = k ? e1[n] : e0[n];
    const float p  = k ? p1[n] : p0[n];

    int myBase = 0;
    #pragma unroll
    for (int q = 0; q < NEXP; ++q) {
        int b = base[r * NEXP + q];          // uniform loads
        if (e == q) myBase = b;
    }

    const unsigned lt = (1u << lane) - 1u;
    int rank = 0;
    #pragma unroll
    for (int q = 0; q < NEXP; ++q) {
        unsigned bal = ballot32(e == q);
        if (e == q) rank = myBase + __popc(bal & lt);
    }

    if (rank < CAP) {
        long idx = (long)n * EC + (long)e * CAP + rank;
        out[CB_OFF + idx]   = p;      // cb_weight
        out[MASK_OFF + idx] = 1.0f;   // sec_mask
    }
}

// ---------------------------------------------------------------------------
extern "C" void kernel_launch(void* const* d_in, const int* in_sizes, int n_in,
                              void* d_out, int out_size, void* d_ws, size_t ws_size,
                              hipStream_t stream) {
    const float* x  = (const float*)d_in[0];   // [4,1024,1024] f32
    const float* wg = (const float*)d_in[1];   // [8,1024] f32
    float* out = (float*)d_out;                // 8 + 2*4096*8*1280 floats

    // Workspace layout (80 KB total)
    int*   e0  = (int*)d_ws;                   // 4096
    int*   e1  = e0 + NTOK;                    // 4096
    float* p0  = (float*)(e1 + NTOK);          // 4096
    float* p1  = p0 + NTOK;                    // 4096
    int*   cnt = (int*)(p1 + NTOK);            // 256*8
    int*   bas = cnt + 256 * NEXP;             // 256*8

    // 1. Zero 335 MB of output — the bandwidth-dominant step.
    zero_out_kernel<<<2048, 256, 0, stream>>>(out, (long)out_size);

    // 2. fp32 WMMA logits + top-2 + softmax (256 waves, one per 16-token tile)
    logits_top2_kernel<<<NTOK / 16, 32, 0, stream>>>(x, wg, e0, e1, p0, p1);

    // 3-5. Parallel k-major rank: histograms -> scan -> scatter
    row_hist_kernel<<<256, 32, 0, stream>>>(e0, e1, cnt);
    scan_kernel<<<1, 256, 0, stream>>>(cnt, bas, out);
    scatter_kernel<<<256, 32, 0, stream>>>(e0, e1, p0, p1, bas, out);
}